// DcorLoss_71141838291708
// MI455X (gfx1250) — compile-verified
//
#include <hip/hip_runtime.h>
#include <math.h>

typedef float v2f __attribute__((ext_vector_type(2)));
typedef float v8f __attribute__((ext_vector_type(8)));

#define NROW 8192
#define DIM  128

// workspace layout (on top of d_ws):
//   double scal[3]          : sum(a*b), sum(a*a), sum(b*b)   (float idx 0..5)
//   float  nx[8192]         : float idx 16
//   float  ny[8192]         : float idx 16+8192
//   float  ra[8192]         : float idx 16+16384   (row sums of a)
//   float  rb[8192]         : float idx 16+24576   (row sums of b)
#define WS_NX 16
#define WS_NY (16 + 8192)
#define WS_RA (16 + 16384)
#define WS_RB (16 + 24576)
#define WS_FLOATS (16 + 32768)

__global__ __launch_bounds__(256) void dcor_zero_ws(float* __restrict__ ws) {
    int i = blockIdx.x * 256 + threadIdx.x;
    if (i < WS_FLOATS) ws[i] = 0.0f;
}

__global__ __launch_bounds__(256) void dcor_row_norms(const float* __restrict__ x,
                                                      const float* __restrict__ y,
                                                      float* __restrict__ nx,
                                                      float* __restrict__ ny) {
    int r = blockIdx.x * 256 + threadIdx.x;   // 8192 threads, one row each
    const float4* xr = (const float4*)(x + (size_t)r * DIM);
    const float4* yr = (const float4*)(y + (size_t)r * DIM);
    float sx = 0.f, sy = 0.f;
#pragma unroll
    for (int t = 0; t < DIM / 4; ++t) {
        float4 a = xr[t];
        float4 b = yr[t];
        sx += a.x * a.x + a.y * a.y + a.z * a.z + a.w * a.w;
        sy += b.x * b.x + b.y * b.y + b.z * b.z + b.w * b.w;
    }
    nx[r] = sx;
    ny[r] = sy;
}

// Each wave computes one 16(i) x 32(j) region of the pairwise-distance matrices
// via V_WMMA_F32_16X16X4_F32 Gram tiles, then reduces everything in-register.
__global__ __launch_bounds__(256) void dcor_gram_tiles(const float* __restrict__ x,
                                                       const float* __restrict__ y,
                                                       const float* __restrict__ nx,
                                                       const float* __restrict__ ny,
                                                       float*  __restrict__ ra,
                                                       float*  __restrict__ rb,
                                                       double* __restrict__ scal) {
    __shared__ double lsd[3];
    if (threadIdx.x < 3) lsd[threadIdx.x] = 0.0;
    __syncthreads();

    const int lane = threadIdx.x & 31;
    const int wave = threadIdx.x >> 5;
    const int task = blockIdx.x * 8 + wave;      // 131072 tasks = 512 (ti) * 256 (tj-pairs)
    const int ti = task >> 8;
    const int tj = task & 255;
    const int i_base = ti * 16;
    const int j_base = tj * 32;

    const int rsel = lane & 15;       // row within the 16-row operand slab
    const int hl   = lane >> 4;       // half: selects K pair (operands) / M+8 (accum)
    const int coff = 2 * hl;          // K column offset for this lane

    const float* xi  = x + (size_t)(i_base + rsel) * DIM + coff;
    const float* xj0 = x + (size_t)(j_base + rsel) * DIM + coff;
    const float* xj1 = xj0 + 16 * DIM;
    const float* yi  = y + (size_t)(i_base + rsel) * DIM + coff;
    const float* yj0 = y + (size_t)(j_base + rsel) * DIM + coff;
    const float* yj1 = yj0 + 16 * DIM;

    v8f gx0 = {0.f,0.f,0.f,0.f,0.f,0.f,0.f,0.f};
    v8f gx1 = {0.f,0.f,0.f,0.f,0.f,0.f,0.f,0.f};
    v8f gy0 = {0.f,0.f,0.f,0.f,0.f,0.f,0.f,0.f};
    v8f gy1 = {0.f,0.f,0.f,0.f,0.f,0.f,0.f,0.f};

#pragma unroll 4
    for (int k = 0; k < DIM; k += 4) {
        v2f a  = *(const v2f*)(xi  + k);
        v2f b0 = *(const v2f*)(xj0 + k);
        v2f b1 = *(const v2f*)(xj1 + k);
        v2f c  = *(const v2f*)(yi  + k);
        v2f d0 = *(const v2f*)(yj0 + k);
        v2f d1 = *(const v2f*)(yj1 + k);
        gx0 = __builtin_amdgcn_wmma_f32_16x16x4_f32(false, a, false, b0, (short)0, gx0, false, false);
        gx1 = __builtin_amdgcn_wmma_f32_16x16x4_f32(false, a, false, b1, (short)0, gx1, false, false);
        gy0 = __builtin_amdgcn_wmma_f32_16x16x4_f32(false, c, false, d0, (short)0, gy0, false, false);
        gy1 = __builtin_amdgcn_wmma_f32_16x16x4_f32(false, c, false, d1, (short)0, gy1, false, false);
    }

    // C/D layout: element p of lane L holds (M = p + 8*(L>>4), N = L&15)
    const int j0 = j_base + rsel;
    const int j1 = j0 + 16;
    const float nxj0 = nx[j0], nxj1 = nx[j1];
    const float nyj0 = ny[j0], nyj1 = ny[j1];

    float sab = 0.f, saa = 0.f, sbb = 0.f;
    float rsa[8], rsb[8];
#pragma unroll
    for (int p = 0; p < 8; ++p) {
        const int i = i_base + p + 8 * hl;
        const float nxi = nx[i], nyi = ny[i];
        float a0 = sqrtf(fmaxf(nxi + nxj0 - 2.f * gx0[p], 0.f));
        float a1 = sqrtf(fmaxf(nxi + nxj1 - 2.f * gx1[p], 0.f));
        float b0 = sqrtf(fmaxf(nyi + nyj0 - 2.f * gy0[p], 0.f));
        float b1 = sqrtf(fmaxf(nyi + nyj1 - 2.f * gy1[p], 0.f));
        if (i == j0) { a0 = 0.f; b0 = 0.f; }   // exact-zero diagonal like reference
        if (i == j1) { a1 = 0.f; b1 = 0.f; }
        sab += a0 * b0 + a1 * b1;
        saa += a0 * a0 + a1 * a1;
        sbb += b0 * b0 + b1 * b1;
        rsa[p] = a0 + a1;
        rsb[p] = b0 + b1;
    }

    // row sums: reduce across the 16 lanes of each half (each row lives in one half)
#pragma unroll
    for (int p = 0; p < 8; ++p) {
        float va = rsa[p], vb = rsb[p];
        for (int m = 1; m < 16; m <<= 1) {
            va += __shfl_xor(va, m, 32);
            vb += __shfl_xor(vb, m, 32);
        }
        rsa[p] = va; rsb[p] = vb;
    }
    if (rsel == 0) {
#pragma unroll
        for (int p = 0; p < 8; ++p) {
            atomicAdd(&ra[i_base + p + 8 * hl], rsa[p]);
            atomicAdd(&rb[i_base + p + 8 * hl], rsb[p]);
        }
    }

    // scalar sums: full-wave butterfly, then f64 LDS accumulation per block
    for (int m = 1; m < 32; m <<= 1) {
        sab += __shfl_xor(sab, m, 32);
        saa += __shfl_xor(saa, m, 32);
        sbb += __shfl_xor(sbb, m, 32);
    }
    if (lane == 0) {
        atomicAdd(&lsd[0], (double)sab);
        atomicAdd(&lsd[1], (double)saa);
        atomicAdd(&lsd[2], (double)sbb);
    }
    __syncthreads();
    if (threadIdx.x == 0) {
        atomicAdd(&scal[0], lsd[0]);
        atomicAdd(&scal[1], lsd[1]);
        atomicAdd(&scal[2], lsd[2]);
    }
}

__global__ __launch_bounds__(256) void dcor_finalize(const float* __restrict__ ra,
                                                     const float* __restrict__ rb,
                                                     const double* __restrict__ scal,
                                                     float* __restrict__ out) {
    __shared__ double red[5][256];
    const int tid = threadIdx.x;
    double daa = 0.0, dbb = 0.0, dab = 0.0, sa = 0.0, sb = 0.0;
    for (int i = tid; i < NROW; i += 256) {
        double A = (double)ra[i], B = (double)rb[i];
        daa += A * A; dbb += B * B; dab += A * B;
        sa += A; sb += B;
    }
    red[0][tid] = daa; red[1][tid] = dbb; red[2][tid] = dab;
    red[3][tid] = sa;  red[4][tid] = sb;
    __syncthreads();
    for (int s = 128; s > 0; s >>= 1) {
        if (tid < s) {
#pragma unroll
            for (int c = 0; c < 5; ++c) red[c][tid] += red[c][tid + s];
        }
        __syncthreads();
    }
    if (tid == 0) {
        const double n = (double)NROW;
        const double inv_n2 = 1.0 / (n * n);
        const double Sa = red[3][0], Sb = red[4][0];
        // sum (HaH)∘(HbH) = Σab − (2/n) rₐ·r_b + SₐS_b/n²
        double sumAB = scal[0] - (2.0 / n) * red[2][0] + Sa * Sb * inv_n2;
        double sumAA = scal[1] - (2.0 / n) * red[0][0] + Sa * Sa * inv_n2;
        double sumBB = scal[2] - (2.0 / n) * red[1][0] + Sb * Sb * inv_n2;
        double dxy = sumAB * inv_n2;
        double dxx = sumAA * inv_n2;
        double dyy = sumBB * inv_n2;
        double dcor = -sqrt(dxy) / sqrt(sqrt(dxx) * sqrt(dyy));
        out[0] = (float)dcor;
    }
}

extern "C" void kernel_launch(void* const* d_in, const int* in_sizes, int n_in,
                              void* d_out, int out_size, void* d_ws, size_t ws_size,
                              hipStream_t stream) {
    const float* x = (const float*)d_in[0];
    const float* y = (const float*)d_in[1];
    float*  out  = (float*)d_out;
    float*  ws   = (float*)d_ws;
    double* scal = (double*)d_ws;          // first 3 doubles (float idx 0..5)
    float*  nx   = ws + WS_NX;
    float*  ny   = ws + WS_NY;
    float*  ra   = ws + WS_RA;
    float*  rb   = ws + WS_RB;

    dcor_zero_ws<<<(WS_FLOATS + 255) / 256, 256, 0, stream>>>(ws);
    dcor_row_norms<<<NROW / 256, 256, 0, stream>>>(x, y, nx, ny);
    // 512 i-tiles * 256 j-pair-tiles = 131072 wave tasks, 8 waves per block
    dcor_gram_tiles<<<131072 / 8, 256, 0, stream>>>(x, y, nx, ny, ra, rb, scal);
    dcor_finalize<<<1, 256, 0, stream>>>(ra, rb, scal, out);
}